// CircularCorrelation_19911468384838
// MI455X (gfx1250) — compile-verified
//
#include <hip/hip_runtime.h>

// CircularCorrelation via radix-64x64 Cooley-Tukey FFT convolution.
// out[n] = IFFT64x64( FFT64x64(a[n]) * FFT64x64(b[n]) ); all 64x64x64 GEMM
// stages run on v_wmma_f32_16x16x4_f32. Memory-bound: ~402 MB -> ~17us floor.
// Global->LDS staging uses the Tensor Data Mover (tensor_load_to_lds) with
// hardware row padding (64 dwords + 4 pad = PITCH 68), pipelined one batch
// ahead and tracked with TENSORcnt.

#define PITCH 68               // 64 + 4: bank-conflict-free rows, 16B-aligned
#define ASZ   (64 * PITCH)     // floats per 64x64 LDS matrix

typedef float v2f __attribute__((ext_vector_type(2)));
typedef float v8f __attribute__((ext_vector_type(8)));
typedef unsigned int v4u __attribute__((ext_vector_type(4)));
typedef int v4i __attribute__((ext_vector_type(4)));
typedef int v8i __attribute__((ext_vector_type(8)));

#if defined(__has_builtin)
#if __has_builtin(__builtin_amdgcn_tensor_load_to_lds) && \
    __has_builtin(__builtin_amdgcn_s_wait_tensorcnt)
#define HAVE_TDM 1
#endif
#endif
#ifndef HAVE_TDM
#define HAVE_TDM 0
#endif

static __device__ __forceinline__ v8f zero8() {
  v8f z = {0.f, 0.f, 0.f, 0.f, 0.f, 0.f, 0.f, 0.f};
  return z;
}

// A-operand fragment (16x4 tile) per ISA layout:
// lanes 0-15: M=lane, VGPR0=K0, VGPR1=K1 ; lanes 16-31: M=lane-16, K2/K3.
static __device__ __forceinline__ v2f ldsA(const float* S, int mtile, int kk, int lane) {
  int row = mtile * 16 + (lane & 15);
  int kb  = kk * 4 + ((lane >> 4) << 1);
  const float* p = S + row * PITCH + kb;
  v2f v; v.x = p[0]; v.y = p[1];
  return v;
}

// B-operand fragment (4x16 tile), mirroring C/D striping:
// VGPR v holds row K=v (lanes 0-15) and K=v+2 (lanes 16-31), N = lane&15.
static __device__ __forceinline__ v2f ldsB(const float* S, int ntile, int kk, int lane) {
  int col = ntile * 16 + (lane & 15);
  int kb  = kk * 4 + ((lane >> 4) << 1);
  v2f v; v.x = S[kb * PITCH + col]; v.y = S[(kb + 1) * PITCH + col];
  return v;
}

static __device__ __forceinline__ v8f wmma4(v2f a, v2f b, v8f c) {
  return __builtin_amdgcn_wmma_f32_16x16x4_f32(false, a, false, b, (short)0, c,
                                               false, false);
}

// C/D tile (16x16) store per ISA layout: VGPR r -> M = r (+8 for lanes 16-31).
static __device__ __forceinline__ void storeC(float* S, int mtile, int ntile,
                                              int lane, v8f c) {
  int col   = ntile * 16 + (lane & 15);
  int rbase = mtile * 16 + ((lane >> 4) << 3);
#pragma unroll
  for (int r = 0; r < 8; ++r) S[(rbase + r) * PITCH + col] = c[r];
}

// Apply twiddle W_4096^(sign * k1*n2) to a C/D tile and store complex result.
static __device__ __forceinline__ void twiddle_store(float* Dr, float* Di,
                                                     int mtile, int ntile,
                                                     int lane, v8f zr, v8f zi,
                                                     float sign) {
  int col   = ntile * 16 + (lane & 15);
  int rbase = mtile * 16 + ((lane >> 4) << 3);
  float n2 = (float)col;
#pragma unroll
  for (int r = 0; r < 8; ++r) {
    int k1 = rbase + r;
    float ang = sign * (6.283185307179586f / 4096.0f) * n2 * (float)k1;
    float s, c;
    __sincosf(ang, &s, &c);
    int idx = k1 * PITCH + col;
    Dr[idx] = zr[r] * c - zi[r] * s;
    Di[idx] = zr[r] * s + zi[r] * c;
  }
}

#if HAVE_TDM
// Issue one TDM descriptor: DMA a 4096-float row (viewed as a 64x64 f32 tile,
// row stride 64) from global into LDS at lds_off, inserting 4 pad DWORDs after
// every 64 DWORDs -> PITCH-68 rows. Tracked by TENSORcnt (wave-0 only).
static __device__ __forceinline__ void tdm_load_row(unsigned lds_off,
                                                    const float* gptr) {
  unsigned long long ga = (unsigned long long)(uintptr_t)gptr;
  v4u g0;
  g0.x = 1u;                                   // count=1, user descriptor
  g0.y = lds_off;                              // lds_addr (bytes)
  g0.z = (unsigned)(ga & 0xffffffffu);         // global_addr[31:0]
  g0.w = (unsigned)((ga >> 32) & 0x01ffffffu)  // global_addr[56:32]
         | (2u << 30);                         // type = 2 ("image")
  v8i g1;
  g1[0] = (2 << 16)    // data_size = 4 bytes
        | (1 << 20)    // pad_enable
        | (5 << 22)    // pad_interval: 2^(5+1) = 64 DWORDs
        | (3 << 25);   // pad_amount: 3+1 = 4 DWORDs
  g1[1] = (64 << 16);  // tensor_dim0 = 64   (bits 79:48)
  g1[2] = (64 << 16);  // tensor_dim1 = 64   (bits 111:80)
  g1[3] = (64 << 16);  // tile_dim0  = 64    (bits 127:112)
  g1[4] = 64;          // tile_dim1  = 64    (bits 143:128); tile_dim2 = 0
  g1[5] = 64;          // tensor_dim0_stride = 64 (bits 207:160)
  g1[6] = 0;
  g1[7] = 0;
  v4i z4 = {0, 0, 0, 0};
#if __clang_major__ >= 23
  v8i z8 = {0, 0, 0, 0, 0, 0, 0, 0};
  __builtin_amdgcn_tensor_load_to_lds(g0, g1, z4, z4, z8, 0);
#else
  __builtin_amdgcn_tensor_load_to_lds(g0, g1, z4, z4, 0);
#endif
}
#else
// Fallback: coalesced 4096-float row -> padded 64x64 LDS matrix.
static __device__ __forceinline__ void load_tile(float* dst, const float* src,
                                                 int tid) {
#pragma unroll
  for (int i = tid; i < 1024; i += 128) {
    const float4 v = reinterpret_cast<const float4*>(src)[i];
    int flat = i << 2;
    int row = flat >> 6, col = flat & 63;
    *reinterpret_cast<float4*>(&dst[row * PITCH + col]) = v;
  }
}
#endif

__global__ void __launch_bounds__(128)
circconv_fft64(const float* __restrict__ Ain, const float* __restrict__ Bin,
               float* __restrict__ Out, int nbatch) {
  extern __shared__ float smem[];
  float* Fr  = smem;            // DFT-64 matrix, real  (symmetric)
  float* Fi  = Fr + ASZ;        // DFT-64 matrix, imag  (e^{-2pi i jk/64})
  float* B0r = Fi + ASZ;        // complex ping buffer
  float* B0i = B0r + ASZ;
  float* B1r = B0i + ASZ;       // complex pong buffer
  float* B1i = B1r + ASZ;
  float* XsA = B1i + ASZ;       // staging buffer for a-rows
  float* XsB = XsA + ASZ;       // staging buffer for b-rows

  const int tid  = threadIdx.x;
  const int lane = tid & 31;
  const int w    = tid >> 5;    // wave id = M-strip id

  // Build DFT-64 matrix once per workgroup.
  for (int i = tid; i < 4096; i += 128) {
    int j = i >> 6, k = i & 63;
    float ang = -(6.283185307179586f / 64.0f) * (float)(j * k);
    float s, c;
    __sincosf(ang, &s, &c);
    Fr[j * PITCH + k] = c;
    Fi[j * PITCH + k] = s;
  }

#if HAVE_TDM
  const unsigned ldsA_off = (unsigned)(uintptr_t)XsA;
  const unsigned ldsB_off = (unsigned)(uintptr_t)XsB;
  // Prologue: DMA this block's first batch (queue = [a, b]).
  if (w == 0 && blockIdx.x < (unsigned)nbatch) {
    tdm_load_row(ldsA_off, Ain + (size_t)blockIdx.x * 4096);
    tdm_load_row(ldsB_off, Bin + (size_t)blockIdx.x * 4096);
  }
#endif
  __syncthreads();

  for (int batch = blockIdx.x; batch < nbatch; batch += gridDim.x) {
    const int nb = batch + gridDim.x;   // this block's next batch
    const bool has_next = nb < nbatch;

    // ---- wait for a-row (oldest of the two outstanding TDM loads) ----
#if HAVE_TDM
    if (w == 0) __builtin_amdgcn_s_wait_tensorcnt((short)1);
#else
    load_tile(XsA, Ain + (size_t)batch * 4096, tid);
    if (has_next) {
      __builtin_prefetch(Ain + (size_t)nb * 4096 + tid * 32, 0, 0);
      __builtin_prefetch(Bin + (size_t)nb * 4096 + tid * 32, 0, 0);
    }
#endif
    __syncthreads();

    // ---- forward stage 1 for a: Y = F * X (X real) ; twiddle ; -> B0 ----
    for (int nt = 0; nt < 4; ++nt) {
      v8f ar = zero8(), ai = zero8();
#pragma unroll
      for (int kk = 0; kk < 16; ++kk) {
        v2f fr = ldsA(Fr, w, kk, lane);
        v2f fi = ldsA(Fi, w, kk, lane);
        v2f xb = ldsB(XsA, nt, kk, lane);
        ar = wmma4(fr, xb, ar);
        ai = wmma4(fi, xb, ai);
      }
      twiddle_store(B0r, B0i, w, nt, lane, ar, ai, -1.0f);
    }
    __syncthreads();

    // ---- forward stage 2 for a: FA = Z * F -> B1 ----
    for (int nt = 0; nt < 4; ++nt) {
      v8f rr = zero8(), ri = zero8(), ir = zero8(), ii = zero8();
#pragma unroll
      for (int kk = 0; kk < 16; ++kk) {
        v2f zr = ldsA(B0r, w, kk, lane);
        v2f zi = ldsA(B0i, w, kk, lane);
        v2f fr = ldsB(Fr, nt, kk, lane);
        v2f fi = ldsB(Fi, nt, kk, lane);
        rr = wmma4(zr, fr, rr);
        ri = wmma4(zr, fi, ri);
        ir = wmma4(zi, fr, ir);
        ii = wmma4(zi, fi, ii);
      }
      storeC(B1r, w, nt, lane, rr - ii);
      storeC(B1i, w, nt, lane, ri + ir);
    }

    // ---- wait for b-row (drain queue); then XsA is free for next batch ----
#if HAVE_TDM
    if (w == 0) __builtin_amdgcn_s_wait_tensorcnt((short)0);
#else
    load_tile(XsB, Bin + (size_t)batch * 4096, tid);
#endif
    __syncthreads();
#if HAVE_TDM
    if (w == 0 && has_next)
      tdm_load_row(ldsA_off, Ain + (size_t)nb * 4096);  // queue = [a']
#endif

    // ---- forward stage 1 for b ----
    for (int nt = 0; nt < 4; ++nt) {
      v8f ar = zero8(), ai = zero8();
#pragma unroll
      for (int kk = 0; kk < 16; ++kk) {
        v2f fr = ldsA(Fr, w, kk, lane);
        v2f fi = ldsA(Fi, w, kk, lane);
        v2f xb = ldsB(XsB, nt, kk, lane);
        ar = wmma4(fr, xb, ar);
        ai = wmma4(fi, xb, ai);
      }
      twiddle_store(B0r, B0i, w, nt, lane, ar, ai, -1.0f);
    }
    __syncthreads();
#if HAVE_TDM
    if (w == 0 && has_next)
      tdm_load_row(ldsB_off, Bin + (size_t)nb * 4096);  // queue = [a', b']
#endif

    // ---- forward stage 2 for b, fused with P = FA .* FB (in-place in B1) ----
    for (int nt = 0; nt < 4; ++nt) {
      v8f rr = zero8(), ri = zero8(), ir = zero8(), ii = zero8();
#pragma unroll
      for (int kk = 0; kk < 16; ++kk) {
        v2f zr = ldsA(B0r, w, kk, lane);
        v2f zi = ldsA(B0i, w, kk, lane);
        v2f fr = ldsB(Fr, nt, kk, lane);
        v2f fi = ldsB(Fi, nt, kk, lane);
        rr = wmma4(zr, fr, rr);
        ri = wmma4(zr, fi, ri);
        ir = wmma4(zi, fr, ir);
        ii = wmma4(zi, fi, ii);
      }
      v8f fbr = rr - ii, fbi = ri + ir;
      int col   = nt * 16 + (lane & 15);
      int rbase = w * 16 + ((lane >> 4) << 3);
#pragma unroll
      for (int r = 0; r < 8; ++r) {
        int idx = (rbase + r) * PITCH + col;
        float far = B1r[idx], fai = B1i[idx];
        B1r[idx] = far * fbr[r] - fai * fbi[r];
        B1i[idx] = far * fbi[r] + fai * fbr[r];
      }
    }
    __syncthreads();

    // ---- inverse stage 1: T = P * conj(F) ; conj twiddle ; -> B0 ----
    for (int nt = 0; nt < 4; ++nt) {
      v8f prfr = zero8(), prfi = zero8(), pifr = zero8(), pifi = zero8();
#pragma unroll
      for (int kk = 0; kk < 16; ++kk) {
        v2f pr = ldsA(B1r, w, kk, lane);
        v2f pi = ldsA(B1i, w, kk, lane);
        v2f fr = ldsB(Fr, nt, kk, lane);
        v2f fi = ldsB(Fi, nt, kk, lane);
        prfr = wmma4(pr, fr, prfr);
        prfi = wmma4(pr, fi, prfi);
        pifr = wmma4(pi, fr, pifr);
        pifi = wmma4(pi, fi, pifi);
      }
      v8f tr = prfr + pifi;  // P * (Fr - i Fi)
      v8f ti = pifr - prfi;
      twiddle_store(B0r, B0i, w, nt, lane, tr, ti, +1.0f);
    }
    __syncthreads();

    // ---- inverse stage 2: out = Re(conj(F)*T)/4096 = (Fr*Tr + Fi*Ti)/4096 ----
    float* outp = Out + (size_t)batch * 4096;
    for (int nt = 0; nt < 4; ++nt) {
      v8f t1 = zero8(), t2 = zero8();
#pragma unroll
      for (int kk = 0; kk < 16; ++kk) {
        v2f fr = ldsA(Fr, w, kk, lane);
        v2f fi = ldsA(Fi, w, kk, lane);
        v2f tr = ldsB(B0r, nt, kk, lane);
        v2f ti = ldsB(B0i, nt, kk, lane);
        t1 = wmma4(fr, tr, t1);
        t2 = wmma4(fi, ti, t2);
      }
      int col   = nt * 16 + (lane & 15);
      int rbase = w * 16 + ((lane >> 4) << 3);
#pragma unroll
      for (int r = 0; r < 8; ++r) {
        outp[(rbase + r) * 64 + col] = (t1[r] + t2[r]) * (1.0f / 4096.0f);
      }
    }
    // next iteration's top-of-loop wait+barrier fences B0/XsA/XsB reuse
    __syncthreads();
  }
}

extern "C" void kernel_launch(void* const* d_in, const int* in_sizes, int n_in,
                              void* d_out, int out_size, void* d_ws, size_t ws_size,
                              hipStream_t stream) {
  const float* a = (const float*)d_in[0];
  const float* b = (const float*)d_in[1];
  float* o = (float*)d_out;
  int nbatch = in_sizes[0] / 4096;
  if (nbatch <= 0) return;
  int nblocks = nbatch < 2048 ? nbatch : 2048;  // >1 batch/block so the TDM
  dim3 grid((unsigned)nblocks), block(128);     // cross-batch prefetch fires
  size_t smem = (size_t)8 * ASZ * sizeof(float);  // ~136 KB of 320 KB WGP LDS
  hipLaunchKernelGGL(circconv_fft64, grid, block, smem, stream, a, b, o, nbatch);
}